// UtilLayer_20169166422902
// MI455X (gfx1250) — compile-verified
//
#include <hip/hip_runtime.h>

// CDNA5 / gfx1250, wave32. FP32 WMMA (V_WMMA_F32_16X16X4_F32) for all GEMMs.
// Weights pre-packed into WMMA B-fragment order: one global_load_b128 per lane
// feeds 4 WMMAs; each wave runs two independent accumulator chains (32 rows).
// ju2 is only ever needed at one dynamically-selected column per edge, so it is
// pre-transposed and consumed as a float4-vectorized per-lane dot product.
typedef __attribute__((ext_vector_type(2))) float v2f;
typedef __attribute__((ext_vector_type(8))) float v8f;

namespace {
constexpr int kB   = 256;                 // subgraphs
constexpr int kNPG = 20;                  // nodes per subgraph
constexpr int kN   = kB * kNPG;           // 5120 nodes
constexpr int kEPG = kNPG * (kNPG - 1);   // 380 edges / subgraph
constexpr int kE   = kB * kEPG;           // 97280 edges
} // namespace

__device__ __forceinline__ v8f wmma_f32(v2f a, v2f b, v8f c) {
  return __builtin_amdgcn_wmma_f32_16x16x4_f32(false, a, false, b, (short)0, c,
                                               false, false);
}

// ---------------------------------------------------------------------------
// Pack W[K][Ncols] (row-major) into B-fragment order:
// float4 index ((nb*KB2 + kb2)*32 + lane) holds, for col = nb*16 + (lane&15),
// ks = 2*(lane>>4):
//   [0] = W[(2*kb2  )*4 + ks + 0][col]   [1] = W[(2*kb2  )*4 + ks + 1][col]
//   [2] = W[(2*kb2+1)*4 + ks + 0][col]   [3] = W[(2*kb2+1)*4 + ks + 1][col]
// ---------------------------------------------------------------------------
__global__ __launch_bounds__(256) void pack_w_kernel(const float* __restrict__ W,
                                                     float* __restrict__ out,
                                                     int K, int Ncols) {
  const int total = (Ncols >> 4) * (K >> 3) * 128;
  const int i = blockIdx.x * blockDim.x + threadIdx.x;
  if (i >= total) return;
  const int q    = i & 3;
  const int lane = (i >> 2) & 31;
  const int t    = i >> 7;               // nb*KB2 + kb2
  const int KB2  = K >> 3;
  const int kb2  = t % KB2;
  const int nb   = t / KB2;
  const int ks   = (lane >> 4) << 1;
  const int n    = lane & 15;
  const int kb   = kb2 * 2 + (q >> 1);
  const int j    = q & 1;
  out[i] = W[(kb * 4 + ks + j) * Ncols + nb * 16 + n];
}

// Transpose ju2_w [128,256] -> [256,128] so a selected column is contiguous.
__global__ __launch_bounds__(256) void transpose_w_kernel(const float* __restrict__ W,
                                                          float* __restrict__ out) {
  const int i = blockIdx.x * blockDim.x + threadIdx.x;   // i < 256*128
  if (i >= 256 * 128) return;
  const int d = i & 127, col = i >> 7;
  out[i] = W[d * 256 + col];
}

// ---------------------------------------------------------------------------
// 32-row tile GEMM: Out[32,Ncols] = act(Xs[32,K] @ W + bias), W pre-packed.
// Two A tiles (rows 0-15, 16-31) share each B fragment -> 2 indep acc chains.
// ---------------------------------------------------------------------------
__device__ __forceinline__ void tile_gemm2(const float* __restrict__ Xs, int xstride,
                                           float* __restrict__ Os, int ostride,
                                           const float4* __restrict__ Wp,
                                           const float* __restrict__ bias,
                                           int K, int Ncols, int lane, bool do_relu) {
  const int m   = lane & 15;
  const int ks  = (lane >> 4) << 1;
  const int n   = lane & 15;
  const int mh  = (lane >> 4) << 3;
  const int KB2 = K >> 3;
  const float* x0 = Xs + m * xstride + ks;
  const float* x1 = Xs + (m + 16) * xstride + ks;
  for (int nb = 0; nb < Ncols; nb += 16) {
    const float bv = bias[nb + n];
    v8f acc0, acc1;
#pragma unroll
    for (int r = 0; r < 8; ++r) { acc0[r] = bv; acc1[r] = bv; }
    const float4* __restrict__ wp = Wp + (nb >> 4) * KB2 * 32 + lane;
    for (int kb2 = 0; kb2 < KB2; ++kb2) {
      const float4 bq = wp[kb2 * 32];           // global_load_b128, coalesced
      v2f b0, b1;
      b0[0] = bq.x; b0[1] = bq.y;
      b1[0] = bq.z; b1[1] = bq.w;
      v2f a0 = *(const v2f*)(x0 + kb2 * 8);     // ds_load_b64
      v2f a1 = *(const v2f*)(x1 + kb2 * 8);
      acc0 = wmma_f32(a0, b0, acc0);
      acc1 = wmma_f32(a1, b0, acc1);
      a0 = *(const v2f*)(x0 + kb2 * 8 + 4);
      a1 = *(const v2f*)(x1 + kb2 * 8 + 4);
      acc0 = wmma_f32(a0, b1, acc0);
      acc1 = wmma_f32(a1, b1, acc1);
    }
#pragma unroll
    for (int r = 0; r < 8; ++r) {
      float v0 = acc0[r], v1 = acc1[r];
      if (do_relu) { v0 = v0 > 0.0f ? v0 : 0.0f; v1 = v1 > 0.0f ? v1 : 0.0f; }
      Os[(r + mh) * ostride + nb + n]        = v0;
      Os[(r + mh + 16) * ostride + nb + n]   = v1;
    }
  }
}

// ---------------------------------------------------------------------------
__global__ __launch_bounds__(256) void init_kernel(float* __restrict__ u_acc,
                                                   float* __restrict__ tsum,
                                                   float* __restrict__ qout) {
  int i = blockIdx.x * blockDim.x + threadIdx.x;
  if (i < kN) { u_acc[i] = 0.0f; tsum[i] = 0.0f; }
  if (i < kB) qout[i] = 0.0f;
}

// ---------------------------------------------------------------------------
// Node MLP (32 nodes / wave): 128 ->relu-> 128 ->relu-> 16; emit
//   pred_indiv[n] = indiv_util[n, acts[n]],  t_i[n] = pred_indiv[n] * weight[n]
// ---------------------------------------------------------------------------
__global__ __launch_bounds__(32) void node_mlp_kernel(
    const float* __restrict__ xin,                                  // [N,128]
    const float4* __restrict__ pw1, const float* __restrict__ b1,
    const float4* __restrict__ pw2, const float* __restrict__ b2,
    const float4* __restrict__ pw3, const float* __restrict__ b3,
    const int* __restrict__ node_batch, const int* __restrict__ acts,
    float* __restrict__ pred_out, float* __restrict__ t_i) {
  __shared__ float Xs[32 * 136];
  __shared__ float Hs[32 * 136];
  __shared__ float Ts[32 * 17];
  const int lane = threadIdx.x;
  const int row0 = blockIdx.x * 32;

  for (int i = lane; i < 32 * 32; i += 32) {     // 32 rows x 128 cols
    const int r = i >> 5, c4 = i & 31;
    float4 v = reinterpret_cast<const float4*>(xin)[(row0 + r) * 32 + c4];
    float* dst = &Xs[r * 136 + c4 * 4];
    dst[0] = v.x; dst[1] = v.y; dst[2] = v.z; dst[3] = v.w;
  }
  __syncthreads();
  tile_gemm2(Xs, 136, Hs, 136, pw1, b1, 128, 128, lane, true);
  __syncthreads();
  tile_gemm2(Hs, 136, Xs, 136, pw2, b2, 128, 128, lane, true);
  __syncthreads();
  tile_gemm2(Xs, 136, Ts, 17, pw3, b3, 128, 16, lane, false);
  __syncthreads();
  {
    const int nidx = row0 + lane;
    const float p = Ts[lane * 17 + acts[nidx]];
    pred_out[nidx] = p;
    float w;
    if (nidx == 0) w = 1.0f;
    else w = (node_batch[nidx] != node_batch[nidx - 1]) ? 1.0f : 0.0f;
    t_i[nidx] = p * w;
  }
}

// ---------------------------------------------------------------------------
// Edge MLP (32 edges / wave): for both feature sets run 192->128->128 (WMMA),
// then only the single needed ju2 column per edge (transposed ju2, float4 dot):
//   s_e = 0.5*( uc[e, k*16+j] + ucr[e, j*16+k] ),  k=acts[src], j=acts[dst]
// atomics: u_acc[dst] += s_e  and (fused scatter) tsum[dst] += t_i[src].
// ---------------------------------------------------------------------------
__global__ __launch_bounds__(32) void edge_mlp_kernel(
    const float* __restrict__ xedge, const float* __restrict__ xrefl,   // [E,192]
    const float4* __restrict__ pw1, const float* __restrict__ b1,       // 192->128
    const float4* __restrict__ pw2, const float* __restrict__ b2,       // 128->128
    const float* __restrict__ w3t, const float* __restrict__ b3,        // [256,128],[256]
    const int* __restrict__ src, const int* __restrict__ dst,
    const int* __restrict__ acts, const float* __restrict__ t_i,
    float* __restrict__ u_acc, float* __restrict__ tsum) {
  __shared__ float Xs[32 * 200];
  __shared__ float Hs[32 * 136];
  const int lane = threadIdx.x;
  const int e0 = blockIdx.x * 32;

  const int e    = e0 + lane;       // lane-private edge
  const int sidx = src[e];
  const int didx = dst[e];
  const int k    = acts[sidx];
  const int j    = acts[didx];

  float myval[2];
#pragma unroll 1
  for (int p = 0; p < 2; ++p) {
    const float* __restrict__ xin = (p == 0) ? xedge : xrefl;
    __syncthreads();                              // Xs reuse across passes
    for (int i = lane; i < 32 * 48; i += 32) {    // 32 rows x 192 cols
      const int rr = i / 48, c4 = i % 48;
      float4 v = reinterpret_cast<const float4*>(xin)[(e0 + rr) * 48 + c4];
      float* dstp = &Xs[rr * 200 + c4 * 4];
      dstp[0] = v.x; dstp[1] = v.y; dstp[2] = v.z; dstp[3] = v.w;
    }
    __syncthreads();
    tile_gemm2(Xs, 200, Hs, 136, pw1, b1, 192, 128, lane, true);
    __syncthreads();
    tile_gemm2(Hs, 136, Xs, 200, pw2, b2, 128, 128, lane, true);
    __syncthreads();
    // selective final column; each lane = one edge row; float4-vectorized
    const int col = (p == 0) ? (k * 16 + j) : (j * 16 + k);
    const float4* __restrict__ g2   = (const float4*)(Xs + lane * 200);
    const float4* __restrict__ wcol = (const float4*)(w3t + col * 128);
    float part = 0.0f;
#pragma unroll 4
    for (int d4 = 0; d4 < 32; ++d4) {
      const float4 x = g2[d4];
      const float4 w = wcol[d4];
      part += x.x * w.x + x.y * w.y + x.z * w.z + x.w * w.w;
    }
    myval[p] = part + b3[col];
  }
  const float s = 0.5f * (myval[0] + myval[1]);
  atomicAdd(&u_acc[didx], s);
  atomicAdd(&tsum[didx], t_i[sidx]);    // fused target_indiv scatter
}

// ---------------------------------------------------------------------------
__global__ __launch_bounds__(256) void finalize_kernel(const float* __restrict__ pred,
                                                       const float* __restrict__ u_acc,
                                                       const float* __restrict__ tsum,
                                                       const float* __restrict__ t_i,
                                                       const int* __restrict__ node_batch,
                                                       float* __restrict__ qout,
                                                       float* __restrict__ treg) {
  int n = blockIdx.x * blockDim.x + threadIdx.x;
  if (n < kN) {
    const float utils = pred[n] + 0.5f * u_acc[n];
    atomicAdd(&qout[node_batch[n]], utils);
    treg[n] = tsum[n] + t_i[n];
  }
}

// ---------------------------------------------------------------------------
extern "C" void kernel_launch(void* const* d_in, const int* in_sizes, int n_in,
                              void* d_out, int out_size, void* d_ws, size_t ws_size,
                              hipStream_t stream) {
  const float* edge_feats = (const float*)d_in[0];
  const float* node_feats = (const float*)d_in[1];
  const float* edge_refl  = (const float*)d_in[2];
  const float* iu1_w = (const float*)d_in[3];  const float* iu1_b = (const float*)d_in[4];
  const float* iu3_w = (const float*)d_in[5];  const float* iu3_b = (const float*)d_in[6];
  const float* iu2_w = (const float*)d_in[7];  const float* iu2_b = (const float*)d_in[8];
  const float* ju1_w = (const float*)d_in[9];  const float* ju1_b = (const float*)d_in[10];
  const float* ju3_w = (const float*)d_in[11]; const float* ju3_b = (const float*)d_in[12];
  const float* ju2_w = (const float*)d_in[13]; const float* ju2_b = (const float*)d_in[14];
  const int* src        = (const int*)d_in[15];
  const int* dst        = (const int*)d_in[16];
  const int* node_batch = (const int*)d_in[17];
  const int* joint_acts = (const int*)d_in[18];

  float* out = (float*)d_out;
  float* qout     = out;                 // [256]
  float* pred_out = out + kB;            // [5120]
  float* treg_out = out + kB + kN;       // [5120]

  float* ws    = (float*)d_ws;
  float* u_acc = ws;                     // [N]
  float* tsum  = ws + kN;                // [N]
  float* t_i   = ws + 2 * kN;            // [N]
  float* p_iu1 = ws + 3 * kN;            // 16384 (128x128 packed)
  float* p_iu3 = p_iu1 + 16384;          // 16384
  float* p_iu2 = p_iu3 + 16384;          // 2048  (128x16 packed)
  float* p_ju1 = p_iu2 + 2048;           // 24576 (192x128 packed)
  float* p_ju3 = p_ju1 + 24576;          // 16384
  float* w3t   = p_ju3 + 16384;          // 32768 (256x128 transposed ju2)
  (void)in_sizes; (void)n_in; (void)out_size; (void)ws_size;

  init_kernel<<<(kN + 255) / 256, 256, 0, stream>>>(u_acc, tsum, qout);
  pack_w_kernel<<<(16384 + 255) / 256, 256, 0, stream>>>(iu1_w, p_iu1, 128, 128);
  pack_w_kernel<<<(16384 + 255) / 256, 256, 0, stream>>>(iu3_w, p_iu3, 128, 128);
  pack_w_kernel<<<(2048  + 255) / 256, 256, 0, stream>>>(iu2_w, p_iu2, 128, 16);
  pack_w_kernel<<<(24576 + 255) / 256, 256, 0, stream>>>(ju1_w, p_ju1, 192, 128);
  pack_w_kernel<<<(16384 + 255) / 256, 256, 0, stream>>>(ju3_w, p_ju3, 128, 128);
  transpose_w_kernel<<<(32768 + 255) / 256, 256, 0, stream>>>(ju2_w, w3t);

  node_mlp_kernel<<<kN / 32, 32, 0, stream>>>(node_feats,
                                              (const float4*)p_iu1, iu1_b,
                                              (const float4*)p_iu3, iu3_b,
                                              (const float4*)p_iu2, iu2_b,
                                              node_batch, joint_acts, pred_out, t_i);

  edge_mlp_kernel<<<kE / 32, 32, 0, stream>>>(edge_feats, edge_refl,
                                              (const float4*)p_ju1, ju1_b,
                                              (const float4*)p_ju3, ju3_b,
                                              w3t, ju2_b,
                                              src, dst, joint_acts, t_i,
                                              u_acc, tsum);

  finalize_kernel<<<(kN + 255) / 256, 256, 0, stream>>>(pred_out, u_acc, tsum, t_i,
                                                        node_batch, qout, treg_out);
}